// GradientLayer_11982958756089
// MI455X (gfx1250) — compile-verified
//
#include <hip/hip_runtime.h>

typedef __attribute__((ext_vector_type(2))) float v2f;
typedef __attribute__((ext_vector_type(8))) float v8f;

#define SMP      8            // samples per workgroup
#define NCH      13           // jet channels: val,x,y,t,xx,xy,yy,xt,yt,xxx,xxy,xyy,yyy
#define NROWS    112          // 14 channels * 8 samples (channel 13 = zero pad)
#define RS       260          // LDS row stride in floats (bank step 4 -> conflict-free A frags)
#define NTHREADS 256          // 8 waves of 32

__device__ inline v8f wmma_f32(v2f a, v2f b, v8f c) {
  // D = A(16x4,f32) * B(4x16,f32) + C(16x16,f32)
  return __builtin_amdgcn_wmma_f32_16x16x4_f32(false, a, false, b, (short)0, c,
                                               false, false);
}

// tanh jet propagation through activation: h = tanh(z) with full chain rule
// over the 13 tracked derivative channels.
__device__ inline void tanh_jet(const float z[NCH], float h[NCH]) {
  float a  = tanhf(z[0]);
  float f1 = 1.0f - a * a;          // f'
  float f2 = -2.0f * a * f1;        // f''
  float f3 = -2.0f * f1 * f1 + 4.0f * a * a * f1;  // f'''
  float zx = z[1], zy = z[2], zt = z[3];
  h[0] = a;
  h[1] = f1 * zx;
  h[2] = f1 * zy;
  h[3] = f1 * zt;
  h[4] = f2 * zx * zx + f1 * z[4];                       // xx
  h[5] = f2 * zx * zy + f1 * z[5];                       // xy
  h[6] = f2 * zy * zy + f1 * z[6];                       // yy
  h[7] = f2 * zx * zt + f1 * z[7];                       // xt
  h[8] = f2 * zy * zt + f1 * z[8];                       // yt
  h[9]  = f3 * zx * zx * zx + 3.0f * f2 * z[4] * zx + f1 * z[9];               // xxx
  h[10] = f3 * zx * zx * zy + f2 * (z[4] * zy + 2.0f * z[5] * zx) + f1 * z[10]; // xxy
  h[11] = f3 * zx * zy * zy + f2 * (z[6] * zx + 2.0f * z[5] * zy) + f1 * z[11]; // xyy
  h[12] = f3 * zy * zy * zy + 3.0f * f2 * z[6] * zy + f1 * z[12];              // yyy
}

__global__ __launch_bounds__(NTHREADS) void pinn_jet_kernel(
    const float* __restrict__ xyt,
    const float* __restrict__ W0, const float* __restrict__ b0,
    const float* __restrict__ W1, const float* __restrict__ b1,
    const float* __restrict__ W2, const float* __restrict__ b2,
    const float* __restrict__ W3, const float* __restrict__ b3,
    const float* __restrict__ W4, const float* __restrict__ b4,
    const float* __restrict__ W5, const float* __restrict__ b5,
    float* __restrict__ out, int N) {
  __shared__ float Hs[NROWS * RS];        // ~114 KB jet buffer
  __shared__ float zf[SMP * 2 * NCH];     // final-layer jets: [s][psi/p][ch]

  const int tid  = threadIdx.x;
  const int lane = tid & 31;
  const int wave = tid >> 5;              // 0..7, owns 32 output neurons
  const int s0   = blockIdx.x * SMP;

  // ---------------- Layer 0: 3 -> 256 (scalar; jet of input is trivial) ----
  {
    const int n  = tid;                   // one neuron per thread, all samples
    const float wx = W0[n], wy = W0[256 + n], wt = W0[512 + n], bb = b0[n];
    for (int s = 0; s < SMP; ++s) {
      const float x = xyt[(s0 + s) * 3 + 0];
      const float y = xyt[(s0 + s) * 3 + 1];
      const float t = xyt[(s0 + s) * 3 + 2];
      float z[NCH], h[NCH];
      z[0] = x * wx + y * wy + t * wt + bb;
      z[1] = wx; z[2] = wy; z[3] = wt;
#pragma unroll
      for (int c = 4; c < NCH; ++c) z[c] = 0.0f;
      tanh_jet(z, h);
#pragma unroll
      for (int c = 0; c < NCH; ++c) Hs[(c * SMP + s) * RS + n] = h[c];
      Hs[(13 * SMP + s) * RS + n] = 0.0f;  // zero the pad channel rows
    }
  }
  __syncthreads();

  // ---------------- Layers 1..4: 256 -> 256 via fp32 WMMA ------------------
  const int arow = lane & 15;             // row within M-tile (A layout)
  const int kofs = (lane >> 4) << 1;      // lanes 0-15: K{0,1}; 16-31: K{2,3}
  const int col0 = wave * 32 + (lane & 15);
  const int col1 = col0 + 16;

#pragma unroll
  for (int L = 0; L < 4; ++L) {
    const float* __restrict__ W = (L == 0) ? W1 : (L == 1) ? W2 : (L == 2) ? W3 : W4;
    const float* __restrict__ bb = (L == 0) ? b1 : (L == 1) ? b2 : (L == 2) ? b3 : b4;

    v8f acc[7][2];
#pragma unroll
    for (int mt = 0; mt < 7; ++mt) {
      v8f zero = {};
      acc[mt][0] = zero;
      acc[mt][1] = zero;
    }

    for (int kc = 0; kc < 64; ++kc) {
      const int k = (kc << 2) + kofs;
      v2f bf0, bf1;                        // B fragments from L2-resident W
      bf0.x = W[k * 256 + col0];
      bf0.y = W[(k + 1) * 256 + col0];
      bf1.x = W[k * 256 + col1];
      bf1.y = W[(k + 1) * 256 + col1];
#pragma unroll
      for (int mt = 0; mt < 7; ++mt) {
        const v2f af = *(const v2f*)&Hs[(mt * 16 + arow) * RS + k];
        acc[mt][0] = wmma_f32(af, bf0, acc[mt][0]);
        acc[mt][1] = wmma_f32(af, bf1, acc[mt][1]);
      }
    }
    __syncthreads();                       // all waves done reading H

    // Scatter Z tiles back into the jet buffer (in place).
    {
      const int nb   = wave * 32 + (lane & 15);
      const int mofs = (lane >> 4) << 3;   // C layout: lanes 16-31 hold M+8
#pragma unroll
      for (int mt = 0; mt < 7; ++mt) {
#pragma unroll
        for (int v = 0; v < 8; ++v) {
          const int r = mt * 16 + mofs + v;
          Hs[r * RS + nb]      = acc[mt][0][v];
          Hs[r * RS + nb + 16] = acc[mt][1][v];
        }
      }
    }
    __syncthreads();

    // Pointwise tanh-jet activation (+ bias on the value channel only).
    {
      const int n = tid;
      const float bn = bb[n];
      for (int s = 0; s < SMP; ++s) {
        float z[NCH], h[NCH];
#pragma unroll
        for (int c = 0; c < NCH; ++c) z[c] = Hs[(c * SMP + s) * RS + n];
        z[0] += bn;
        tanh_jet(z, h);
#pragma unroll
        for (int c = 0; c < NCH; ++c) Hs[(c * SMP + s) * RS + n] = h[c];
      }
    }
    __syncthreads();
  }

  // ---------------- Layer 5: 256 -> 2 (psi, p) jets, scalar dots -----------
  if (tid < SMP * 2 * NCH) {              // 208 dot products of length 256
    const int c = tid % NCH;
    const int j = (tid / NCH) & 1;        // 0 = psi, 1 = p
    const int s = tid / (2 * NCH);
    const float* hrow = &Hs[(c * SMP + s) * RS];
    float acc = 0.0f;
    int kk = (lane * 8) & 255;            // stagger start to spread LDS banks
    for (int k = 0; k < 256; ++k) {
      acc += hrow[kk] * W5[kk * 2 + j];
      kk = (kk + 1) & 255;
    }
    zf[tid] = acc;
  }
  __syncthreads();

  // ---------------- Assemble the 16 output fields --------------------------
  if (tid < SMP * 16) {
    const int s = tid >> 4;
    const int o = tid & 15;
    const float* S = &zf[(s * 2 + 0) * NCH];   // psi jet
    const float* P = &zf[(s * 2 + 1) * NCH];   // p jet
    float val;
    switch (o) {
      case 0:  val = S[0] + b5[0]; break;      // psi
      case 1:  val = P[0] + b5[1]; break;      // p
      case 2:  val = P[1];  break;             // p_x
      case 3:  val = P[2];  break;             // p_y
      case 4:  val = S[2];  break;             // u  = psi_y
      case 5:  val = S[5];  break;             // u_x = psi_xy
      case 6:  val = S[6];  break;             // u_y = psi_yy
      case 7:  val = S[10]; break;             // u_xx = psi_xxy
      case 8:  val = S[12]; break;             // u_yy = psi_yyy
      case 9:  val = -S[1]; break;             // v  = -psi_x
      case 10: val = -S[4]; break;             // v_x = -psi_xx
      case 11: val = -S[5]; break;             // v_y = -psi_xy
      case 12: val = -S[9]; break;             // v_xx = -psi_xxx
      case 13: val = -S[11]; break;            // v_yy = -psi_xyy
      case 14: val = S[8];  break;             // u_t = psi_yt
      default: val = -S[7]; break;             // v_t = -psi_xt
    }
    out[o * N + s0 + s] = val;
  }
}

extern "C" void kernel_launch(void* const* d_in, const int* in_sizes, int n_in,
                              void* d_out, int out_size, void* d_ws, size_t ws_size,
                              hipStream_t stream) {
  const float* xyt = (const float*)d_in[0];
  const float* W0 = (const float*)d_in[1];
  const float* b0 = (const float*)d_in[2];
  const float* W1 = (const float*)d_in[3];
  const float* b1 = (const float*)d_in[4];
  const float* W2 = (const float*)d_in[5];
  const float* b2 = (const float*)d_in[6];
  const float* W3 = (const float*)d_in[7];
  const float* b3 = (const float*)d_in[8];
  const float* W4 = (const float*)d_in[9];
  const float* b4 = (const float*)d_in[10];
  const float* W5 = (const float*)d_in[11];
  const float* b5 = (const float*)d_in[12];
  float* out = (float*)d_out;

  const int N = in_sizes[0] / 3;
  const int blocks = N / SMP;
  pinn_jet_kernel<<<blocks, NTHREADS, 0, stream>>>(
      xyt, W0, b0, W1, b1, W2, b2, W3, b3, W4, b4, W5, b5, out, N);
}